// VAELoss_22222160789852
// MI455X (gfx1250) — compile-verified
//
#include <hip/hip_runtime.h>

typedef __attribute__((ext_vector_type(2))) float v2f;
typedef __attribute__((ext_vector_type(8))) float v8f;

#define BATCHES        8
#define NPTS           4096
#define TILE           16
#define NTILES         (NPTS / TILE)        // 256 column tiles
#define ROWBLOCKS      (NPTS / TILE)        // 256 row blocks
#define BLOCKS_PER_PASS (BATCHES * ROWBLOCKS)   // 2048
#define TOTAL_BLOCKS   (2 * BLOCKS_PER_PASS)    // 4096

// One wave (32 lanes) per block. Each block owns (pass, batch, 16-row block),
// sweeps all 256 column tiles. Per tile: one V_WMMA_F32_16X16X4_F32 computes
// the full 16x16 tile of squared distances via  d2 = (-2P)·Q^T + (|p|^2+|q|^2).
// Column-point loads are fully uniform per half-wave mirror (all 3 coords per
// lane, branch-free selects) so the inner loop has no EXEC divergence and no
// cross-lane (LDS) traffic.
__global__ __launch_bounds__(32)
void chamfer_wmma_kernel(const float* __restrict__ pred,
                         const float* __restrict__ targ,
                         float* __restrict__ partial)
{
    const int  lane = threadIdx.x;       // 0..31
    const bool ishi = lane >= 16;        // high half: K=2,3 / rows 8..15
    const int  idx  = lane & 15;         // point index within tile / N index

    const int bid  = blockIdx.x;
    const int pass = bid / BLOCKS_PER_PASS;          // 0: pred rows, 1: target rows
    const int rem  = bid % BLOCKS_PER_PASS;
    const int b    = rem / ROWBLOCKS;
    const int rb   = rem % ROWBLOCKS;

    const float* rows = (pass == 0) ? pred : targ;
    const float* cols = (pass == 0) ? targ : pred;
    rows += (size_t)b * NPTS * 3;
    cols += (size_t)b * NPTS * 3;

    // ---- Load this block's 16 row points (lane & lane+16 mirror point idx) ----
    const int rp = rb * TILE + idx;
    const float p0 = rows[rp * 3 + 0];
    const float p1 = rows[rp * 3 + 1];
    const float p2 = rows[rp * 3 + 2];

    // A-matrix (16x4): row m = -2 * point m, K=3 padded with zero.
    // Lanes 0-15 hold K=0,1 ; lanes 16-31 hold K=2,3.
    v2f a;
    a.x = -2.0f * (ishi ? p2 : p0);
    a.y = ishi ? 0.0f : (-2.0f * p1);

    // |P_m|^2, broadcast so each lane holds norms for its C/D half's rows
    // (one-time setup; cross-lane reads are fine here).
    const float pn2_self = p0 * p0 + p1 * p1 + p2 * p2;
    float pn2v[8];
#pragma unroll
    for (int v = 0; v < 8; ++v)
        pn2v[v] = __shfl(pn2_self, v + (ishi ? 8 : 0), 32);  // row m = v + 8*hi

    v8f acc;
#pragma unroll
    for (int v = 0; v < 8; ++v) acc[v] = 3.4e38f;

#pragma unroll 4
    for (int j = 0; j < NTILES; ++j) {
        // Every lane loads all 3 coords of column point j*16+idx (halves mirror):
        // unconditional, coalesced, L2-resident -> no divergent loads.
        const int cbase = (j * TILE + idx) * 3;
        const float c0 = cols[cbase + 0];
        const float c1 = cols[cbase + 1];
        const float c2 = cols[cbase + 2];

        // B-matrix (4x16): lanes 0-15 carry K=0,1 ; lanes 16-31 carry K=2,3(=0).
        // Branch-free selects (v_cndmask on a loop-invariant VCC).
        v2f bm;
        bm.x = ishi ? c2 : c0;
        bm.y = ishi ? 0.0f : c1;

        // |Q_n|^2 computed entirely in-lane -> no in-loop shuffle/LDS.
        const float qn2 = c0 * c0 + c1 * c1 + c2 * c2;

        // C(m,n) = |P_m|^2 + |Q_n|^2  (per C/D VGPR layout)
        v8f c;
#pragma unroll
        for (int v = 0; v < 8; ++v) c[v] = pn2v[v] + qn2;

        // D = A x B + C  ->  full 16x16 tile of squared distances
        v8f d = __builtin_amdgcn_wmma_f32_16x16x4_f32(
            false, a, false, bm, (short)0, c, false, false);

#pragma unroll
        for (int v = 0; v < 8; ++v) acc[v] = fminf(acc[v], d[v]);
    }

    // Row-min: butterfly min across the 16 N-lanes of each half (one-time).
#pragma unroll
    for (int w = 1; w < 16; w <<= 1) {
#pragma unroll
        for (int v = 0; v < 8; ++v)
            acc[v] = fminf(acc[v], __shfl_xor(acc[v], w, 32));
    }

    // Sum sqrt(min d^2) over this half's 8 rows, combine halves, write partial.
    float s = 0.0f;
#pragma unroll
    for (int v = 0; v < 8; ++v)
        s += sqrtf(fmaxf(acc[v], 0.0f));   // clamp: norm-trick cancellation
    s += __shfl_xor(s, 16, 32);

    if (lane == 0) partial[bid] = s;
}

// Deterministic single-block tree reduction of the 4096 partials.
__global__ __launch_bounds__(256)
void chamfer_reduce_kernel(const float* __restrict__ partial,
                           float* __restrict__ out)
{
    __shared__ float sm[256];
    float s = 0.0f;
    for (int i = threadIdx.x; i < TOTAL_BLOCKS; i += 256) s += partial[i];
    sm[threadIdx.x] = s;
    __syncthreads();
    for (int off = 128; off > 0; off >>= 1) {
        if (threadIdx.x < off) sm[threadIdx.x] += sm[threadIdx.x + off];
        __syncthreads();
    }
    if (threadIdx.x == 0)
        out[0] = sm[0] * (1.0f / (float)(BATCHES * NPTS));  // CD_COEFF = 1
}

extern "C" void kernel_launch(void* const* d_in, const int* in_sizes, int n_in,
                              void* d_out, int out_size, void* d_ws, size_t ws_size,
                              hipStream_t stream)
{
    const float* pred = (const float*)d_in[0];   // (8,4096,3) f32
    const float* targ = (const float*)d_in[1];   // (8,4096,3) f32
    float* out     = (float*)d_out;              // scalar f32
    float* partial = (float*)d_ws;               // 4096 floats of scratch

    chamfer_wmma_kernel<<<TOTAL_BLOCKS, 32, 0, stream>>>(pred, targ, partial);
    chamfer_reduce_kernel<<<1, 256, 0, stream>>>(partial, out);

    (void)in_sizes; (void)n_in; (void)out_size; (void)ws_size;
}